// ConnectionProductBlock_36876589203912
// MI455X (gfx1250) — compile-verified
//
#include <hip/hip_runtime.h>

// Shapes fixed by the reference: am_out (16,8,56,56) f32, first_out (16,64,56,56) f32
// out (16, 512, 56, 56) f32 ; out[b, c*64+k, h, w] = am[b,c,h,w] * first[b,k,h,w]
constexpr int B_  = 16;
constexpr int C_  = 8;
constexpr int K_  = 64;
constexpr int HW  = 56 * 56;   // 3136 floats per plane
constexpr int HW4 = HW / 4;    // 784 float4 per plane
constexpr int NPLANES = B_ * C_ * K_;  // 8192 output planes
constexpr int THREADS = 256;           // 8 waves (wave32)

typedef float f4  __attribute__((ext_vector_type(4)));
typedef int   v4i __attribute__((ext_vector_type(4)));

// gfx1250 async global->LDS path (ASYNCcnt), used when toolchain exposes it.
#if defined(__has_builtin)
#if __has_builtin(__builtin_amdgcn_global_load_async_to_lds_b128) && \
    __has_builtin(__builtin_amdgcn_s_wait_asynccnt)
#define USE_ASYNC_LDS 1
#endif
#endif

__global__ __launch_bounds__(THREADS)
void plane_product_kernel(const float* __restrict__ am,
                          const float* __restrict__ fo,
                          float* __restrict__ out) {
    const int p   = blockIdx.x;        // 0..8191
    const int b   = p >> 9;            // / (C*K)
    const int ck  = p & (C_ * K_ - 1);
    const int c   = ck >> 6;           // / K
    const int k   = ck & (K_ - 1);
    const int tid = threadIdx.x;

    const f4* __restrict__ amv = (const f4*)(am + (size_t)(b * C_ + c) * HW);
    const f4* __restrict__ fov = (const f4*)(fo + (size_t)(b * K_ + k) * HW);
    f4* __restrict__       ov  = (f4*)(out + (size_t)p * HW);

#if defined(USE_ASYNC_LDS)
    // Stage both reused input planes into LDS with async B128 copies
    // (gfx1250 GLOBAL_LOAD_ASYNC_TO_LDS_B128, tracked by ASYNCcnt).
    __shared__ f4 s_am[HW4];
    __shared__ f4 s_fo[HW4];
    for (int i = tid; i < HW4; i += THREADS) {
        __builtin_amdgcn_global_load_async_to_lds_b128(
            (v4i*)(amv + i), (v4i*)(s_am + i), 0, 0);
        __builtin_amdgcn_global_load_async_to_lds_b128(
            (v4i*)(fov + i), (v4i*)(s_fo + i), 0, 0);
    }
    __builtin_amdgcn_s_wait_asynccnt(0);
    __syncthreads();
    for (int i = tid; i < HW4; i += THREADS) {
        f4 r = s_am[i] * s_fo[i];
        __builtin_nontemporal_store(r, ov + i);   // 103 MB write-once stream -> NT
    }
#else
    // Fallback: direct B128 loads (planes are L2-resident after first touch).
    for (int i = tid; i < HW4; i += THREADS) {
        f4 a = amv[i];
        f4 f = fov[i];
        f4 r = a * f;
        __builtin_nontemporal_store(r, ov + i);
    }
#endif
}

extern "C" void kernel_launch(void* const* d_in, const int* in_sizes, int n_in,
                              void* d_out, int out_size, void* d_ws, size_t ws_size,
                              hipStream_t stream) {
    const float* am = (const float*)d_in[0];   // am_out    (16,8,56,56)
    const float* fo = (const float*)d_in[1];   // first_out (16,64,56,56)
    float* out      = (float*)d_out;           // (16,512,56,56)
    (void)in_sizes; (void)n_in; (void)out_size; (void)d_ws; (void)ws_size;

    plane_product_kernel<<<dim3(NPLANES), dim3(THREADS), 0, stream>>>(am, fo, out);
}